// EnhancedGraphGenerator_81784767250569
// MI455X (gfx1250) — compile-verified
//
#include <hip/hip_runtime.h>
#include <hip/hip_bf16.h>

#define NN 1024   // nodes
#define FF 512    // input features
#define HH 64     // hidden

typedef float v2f __attribute__((ext_vector_type(2)));
typedef float v8f __attribute__((ext_vector_type(8)));

// ---------------------------------------------------------------------------
// Generic fp32 WMMA GEMM:  C[M x NC] = epilogue(A[M x K] @ B[K x NC] + bias)
// One wave (32 lanes) computes one 16x16 tile via V_WMMA_F32_16X16X4_F32.
//   A fragment (16x4):  lanes 0-15 hold row M=lane  K={k,k+1};
//                       lanes 16-31 hold row M=lane-16 K={k+2,k+3}
//   B fragment (4x16):  VGPR0 = rows k (lanes 0-15) / k+2 (lanes 16-31),
//                       VGPR1 = rows k+1 / k+3, col N = lane%16
//   D (16x16 f32):      VGPR r -> row r (lanes 0-15) / row r+8 (lanes 16-31)
// split_cols>0 maps output column blocks onto stacked B parts (for Wa1[2H,H]).
// ---------------------------------------------------------------------------
__global__ __launch_bounds__(32) void k_gemm_wmma_f32(
    const float* __restrict__ A, int lda,
    const float* __restrict__ B, int ldb, int split_cols,
    const float* __restrict__ bias,
    float* __restrict__ C, int ldc,
    int K, int do_relu)
{
  const int i0 = blockIdx.x << 4;
  const int j0 = blockIdx.y << 4;

  const float* Bp = B;
  int jb = j0;
  if (split_cols > 0) {                       // B stacked parts along K-rows
    Bp += (size_t)(j0 / split_cols) * (size_t)K * (size_t)ldb;
    jb  = j0 % split_cols;
  }

  const int lane = threadIdx.x & 31;
  const int half = lane >> 4;                 // 0: K={0,1}, 1: K={2,3}
  const int l16  = lane & 15;

  const float* Arow = A + (size_t)(i0 + l16) * lda;
  const float* Bcol = Bp + jb + l16;

  v8f acc = {};
  for (int k = 0; k < K; k += 4) {
    const int kk = k + (half << 1);
    v2f a, b;
    a.x = Arow[kk];
    a.y = Arow[kk + 1];
    b.x = Bcol[(size_t)kk * ldb];
    b.y = Bcol[(size_t)(kk + 1) * ldb];
    acc = __builtin_amdgcn_wmma_f32_16x16x4_f32(
        false, a, false, b, (short)0, acc, false, false);
  }

  const float bv = bias ? bias[j0 + l16] : 0.0f;
#pragma unroll
  for (int r = 0; r < 8; ++r) {
    float v = acc[r] + bv;
    if (do_relu) v = fmaxf(v, 0.0f);
    C[(size_t)(i0 + r + (half << 3)) * ldc + (j0 + l16)] = v;
  }
}

// ---------------------------------------------------------------------------
// Pairwise scores: one 16x16 (i,j) tile per 256-thread block.
// Each thread computes both s[i,j] and s[j,i] contributions so the
// symmetrization needs no transpose pass.  Rows padded to 65 floats to avoid
// stride-64 LDS bank conflicts (64 banks).
// ---------------------------------------------------------------------------
__global__ __launch_bounds__(256) void k_pairwise(
    const float* __restrict__ ei, const float* __restrict__ ej, int lde,
    const float* __restrict__ ba1, const float* __restrict__ wa2,
    const float* __restrict__ ba2, const float* __restrict__ temperature,
    float* __restrict__ adj, float* __restrict__ partials)
{
  __shared__ float sEiI[16][HH + 1];
  __shared__ float sEjI[16][HH + 1];
  __shared__ float sEiJ[16][HH + 1];
  __shared__ float sEjJ[16][HH + 1];
  __shared__ float sBa[HH], sWa[HH];
  __shared__ float red[256];

  const int i0  = blockIdx.y << 4;
  const int j0  = blockIdx.x << 4;
  const int tid = threadIdx.x;

  for (int idx = tid; idx < 16 * HH; idx += 256) {
    const int r = idx >> 6;         // / 64
    const int c = idx & 63;
    sEiI[r][c] = ei[(size_t)(i0 + r) * lde + c];
    sEjI[r][c] = ej[(size_t)(i0 + r) * lde + c];
    sEiJ[r][c] = ei[(size_t)(j0 + r) * lde + c];
    sEjJ[r][c] = ej[(size_t)(j0 + r) * lde + c];
  }
  if (tid < HH) { sBa[tid] = ba1[tid]; sWa[tid] = wa2[tid]; }
  __syncthreads();

  const int ti = tid >> 4;
  const int tj = tid & 15;

  float s1 = 0.0f, s2 = 0.0f;
#pragma unroll 8
  for (int h = 0; h < HH; ++h) {
    const float bb = sBa[h];
    const float ww = sWa[h];
    const float z1 = sEiI[ti][h] + sEjJ[tj][h] + bb;   // score(i,j) term
    const float z2 = sEiJ[tj][h] + sEjI[ti][h] + bb;   // score(j,i) term
    s1 += fmaxf(z1, 0.0f) * ww;
    s2 += fmaxf(z2, 0.0f) * ww;
  }

  float t = *temperature;
  t = fminf(fmaxf(t, 0.1f), 5.0f);
  const float bo  = *ba2;
  const float sym = 0.5f * ((s1 + bo) + (s2 + bo)) / t;
  const float a   = 1.0f / (1.0f + __expf(-sym));      // sigmoid; a > 0 so |a| == a

  adj[(size_t)(i0 + ti) * NN + (j0 + tj)] = a;

  // deterministic per-block partial sum for the sparsity loss
  red[tid] = a;
  __syncthreads();
#pragma unroll
  for (int s = 128; s > 0; s >>= 1) {
    if (tid < s) red[tid] += red[tid + s];
    __syncthreads();
  }
  if (tid == 0) partials[blockIdx.y * gridDim.x + blockIdx.x] = red[0];
}

// ---------------------------------------------------------------------------
// Deterministic final reduction of block partials -> sparsity loss scalar.
// ---------------------------------------------------------------------------
__global__ __launch_bounds__(256) void k_reduce(
    const float* __restrict__ partials, int n, float* __restrict__ out,
    float scale)
{
  __shared__ float red[256];
  float s = 0.0f;
  for (int i = threadIdx.x; i < n; i += 256) s += partials[i];
  red[threadIdx.x] = s;
  __syncthreads();
#pragma unroll
  for (int st = 128; st > 0; st >>= 1) {
    if (threadIdx.x < st) red[threadIdx.x] += red[threadIdx.x + st];
    __syncthreads();
  }
  if (threadIdx.x == 0) out[0] = red[0] * scale;
}

extern "C" void kernel_launch(void* const* d_in, const int* in_sizes, int n_in,
                              void* d_out, int out_size, void* d_ws, size_t ws_size,
                              hipStream_t stream) {
  const float* X    = (const float*)d_in[0];   // [1024, 512]
  const float* W1   = (const float*)d_in[1];   // [512, 64]
  const float* b1   = (const float*)d_in[2];   // [64]
  const float* W2   = (const float*)d_in[3];   // [64, 64]
  const float* b2   = (const float*)d_in[4];   // [64]
  const float* Wa1  = (const float*)d_in[5];   // [128, 64]
  const float* ba1  = (const float*)d_in[6];   // [64]
  const float* wa2  = (const float*)d_in[7];   // [64]
  const float* ba2  = (const float*)d_in[8];   // scalar
  const float* temp = (const float*)d_in[9];   // scalar

  float* adj  = (float*)d_out;                 // [1024, 1024]
  float* loss = adj + (size_t)NN * NN;         // scalar at the end

  // workspace layout (floats)
  float* t        = (float*)d_ws;              // [1024, 64]
  float* h        = t + NN * HH;               // [1024, 64]
  float* e        = h + NN * HH;               // [1024, 128] = [ei | ej]
  float* partials = e + NN * 2 * HH;           // [64*64]

  const dim3 wblk(32);
  // t = relu(X @ W1 + b1)
  k_gemm_wmma_f32<<<dim3(NN / 16, HH / 16), wblk, 0, stream>>>(
      X, FF, W1, HH, 0, b1, t, HH, FF, 1);
  // h = t @ W2 + b2
  k_gemm_wmma_f32<<<dim3(NN / 16, HH / 16), wblk, 0, stream>>>(
      t, HH, W2, HH, 0, b2, h, HH, HH, 0);
  // e = [h @ Wa1[:64] | h @ Wa1[64:]]   (split_cols maps col block -> Wa1 part)
  k_gemm_wmma_f32<<<dim3(NN / 16, (2 * HH) / 16), wblk, 0, stream>>>(
      h, HH, Wa1, HH, HH, nullptr, e, 2 * HH, HH, 0);

  // pairwise scores -> adjacency + per-block loss partials
  k_pairwise<<<dim3(NN / 16, NN / 16), 256, 0, stream>>>(
      e, e + HH, 2 * HH, ba1, wa2, ba2, temp, adj, partials);

  // loss = 0.01 * mean(adjacency)
  k_reduce<<<1, 256, 0, stream>>>(
      partials, (NN / 16) * (NN / 16), loss, 0.01f / ((float)NN * (float)NN));
}